// DistanceOverTime_1821066134270
// MI455X (gfx1250) — compile-verified
//
#include <hip/hip_runtime.h>

typedef float v2f __attribute__((ext_vector_type(2)));
typedef float v8f __attribute__((ext_vector_type(8)));

#define BATCH 8
#define NPTS  2048
#define DIM   64
#define NT    (NPTS / 16)   // 128 tiles per matrix dimension
#define KSTEPS (DIM / 4)    // 16 f32 WMMAs (K=4 each) per matrix per tile

// ---------------------------------------------------------------------------
// Kernel 1: per-row squared norms for both tensors.
// ---------------------------------------------------------------------------
__global__ void norms_kernel(const float* __restrict__ xo,
                             const float* __restrict__ xt,
                             float* __restrict__ sqo,
                             float* __restrict__ sqt) {
    int row = blockIdx.x * blockDim.x + threadIdx.x;   // 0 .. BATCH*NPTS-1
    if (row >= BATCH * NPTS) return;
    const float* po = xo + (size_t)row * DIM;
    const float* pt = xt + (size_t)row * DIM;
    float so = 0.f, st = 0.f;
#pragma unroll
    for (int d = 0; d < DIM; ++d) {
        so += po[d] * po[d];
        st += pt[d] * pt[d];
    }
    sqo[row] = so;
    sqt[row] = st;
}

// ---------------------------------------------------------------------------
// Kernel 2: fused Gram-tile (f32 WMMA) -> distances -> |diff| -> block partial.
// One block per (batch, i-tile); 8 waves split the j-tile range [ti, NT).
// Symmetry: tj > ti tiles weighted 2, diagonal tile weighted 1.
//
// Schedule (enforced with sched_barrier): per j-tile, ALL 32 B-fragment
// loads issue as one clause (32 loads in flight -> one L2 latency exposed),
// then 32 WMMAs run back-to-back in 4 independent accumulator chains.
// amdgpu_waves_per_eu(1) lifts the register ceiling so the B-array stays
// resident instead of being streamed through one register pair.
// ---------------------------------------------------------------------------
__global__ void __launch_bounds__(256)
__attribute__((amdgpu_waves_per_eu(1)))
dist_tile_kernel(const float* __restrict__ xo,
                 const float* __restrict__ xt,
                 const float* __restrict__ sqo,
                 const float* __restrict__ sqt,
                 float* __restrict__ partials) {
    __shared__ float wsum[8];

    const int lane = threadIdx.x & 31;
    const int wave = threadIdx.x >> 5;
    const int b    = blockIdx.x / NT;
    const int ti   = blockIdx.x % NT;

    // f32 WMMA 16x16x4 A/B fragment addressing (per ISA layout):
    //   lanes 0-15 : M/N = lane,     K = {0,1} in VGPR0/1
    //   lanes 16-31: M/N = lane-16,  K = {2,3} in VGPR0/1
    const int m    = lane & 15;
    const int koff = (lane >> 4) << 1;     // 0 or 2
    const int hi8  = (lane >> 4) << 3;     // C/D layout: upper lanes hold M+8

    const size_t xbase = (size_t)b * NPTS * DIM;
    const int    row_i = ti * 16 + m;

    // A-fragments for both tensors, all 16 K-steps, held in registers.
    v2f ao[KSTEPS], at[KSTEPS];
#pragma unroll
    for (int s = 0; s < KSTEPS; ++s) {
        ao[s] = *(const v2f*)(xo + xbase + (size_t)row_i * DIM + 4 * s + koff);
        at[s] = *(const v2f*)(xt + xbase + (size_t)row_i * DIM + 4 * s + koff);
    }

    // Row norms for the 8 output rows this lane owns (C/D layout: M = r + hi8).
    float rno[8], rnt[8];
#pragma unroll
    for (int r = 0; r < 8; ++r) {
        rno[r] = sqo[b * NPTS + ti * 16 + r + hi8];
        rnt[r] = sqt[b * NPTS + ti * 16 + r + hi8];
    }

    float sum = 0.f;

    for (int tj = ti + wave; tj < NT; tj += 8) {       // wave-uniform loop
        const int   col = tj * 16 + m;                 // B-frag: N = lane&15

        // ---- Load phase: all B-fragments + col norms up front. ----
        const float* pbo = xo + xbase + (size_t)col * DIM + koff;
        const float* pbt = xt + xbase + (size_t)col * DIM + koff;

        v2f bo[KSTEPS], bt[KSTEPS];
#pragma unroll
        for (int s = 0; s < KSTEPS; ++s) {
            bo[s] = *(const v2f*)(pbo + 4 * s);
            bt[s] = *(const v2f*)(pbt + 4 * s);
        }
        const float cno = sqo[b * NPTS + col];
        const float cnt = sqt[b * NPTS + col];

        // Prefetch next j-tile's rows into the near cache (global_prefetch_b8).
        if (tj + 8 < NT) {
            const float* nbo = pbo + 8 * 16 * DIM;
            const float* nbt = pbt + 8 * 16 * DIM;
            __builtin_prefetch(nbo, 0, 3);
            __builtin_prefetch(nbo + 32, 0, 3);   // second 128B line of the row
            __builtin_prefetch(nbt, 0, 3);
            __builtin_prefetch(nbt + 32, 0, 3);
        }

        // Keep loads on this side; WMMAs on the other side.
        __builtin_amdgcn_sched_barrier(0);

        // ---- Compute phase: 4 independent accumulator chains. ----
        v8f acc_o0 = {}, acc_o1 = {};
        v8f acc_t0 = {}, acc_t1 = {};
#pragma unroll
        for (int s = 0; s < KSTEPS; s += 2) {
            acc_o0 = __builtin_amdgcn_wmma_f32_16x16x4_f32(
                false, ao[s],     false, bo[s],     (short)0, acc_o0, false, false);
            acc_t0 = __builtin_amdgcn_wmma_f32_16x16x4_f32(
                false, at[s],     false, bt[s],     (short)0, acc_t0, false, false);
            acc_o1 = __builtin_amdgcn_wmma_f32_16x16x4_f32(
                false, ao[s + 1], false, bo[s + 1], (short)0, acc_o1, false, false);
            acc_t1 = __builtin_amdgcn_wmma_f32_16x16x4_f32(
                false, at[s + 1], false, bt[s + 1], (short)0, acc_t1, false, false);
        }

        const float w = (tj == ti) ? 1.0f : 2.0f;
#pragma unroll
        for (int r = 0; r < 8; ++r) {
            float go  = acc_o0[r] + acc_o1[r];
            float gt  = acc_t0[r] + acc_t1[r];
            float d2o = rno[r] + cno - 2.0f * go;
            float d2t = rnt[r] + cnt - 2.0f * gt;
            float dgo = sqrtf(fmaxf(d2o, 0.0f));
            float dgt = sqrtf(fmaxf(d2t, 0.0f));
            sum += w * fabsf(dgo - dgt);
        }
    }

    // Deterministic wave32 reduction (fixed xor-shuffle order).
#pragma unroll
    for (int off = 16; off >= 1; off >>= 1)
        sum += __shfl_xor(sum, off, 32);

    if (lane == 0) wsum[wave] = sum;
    __syncthreads();
    if (threadIdx.x == 0) {
        float t = 0.f;
#pragma unroll
        for (int w2 = 0; w2 < 8; ++w2) t += wsum[w2];
        partials[blockIdx.x] = t;
    }
}

// ---------------------------------------------------------------------------
// Kernel 3: deterministic final reduction of the 1024 block partials + mean.
// ---------------------------------------------------------------------------
__global__ void reduce_kernel(const float* __restrict__ partials,
                              float* __restrict__ out) {
    __shared__ float sh[256];
    float s = 0.f;
    for (int i = threadIdx.x; i < BATCH * NT; i += 256) s += partials[i];
    sh[threadIdx.x] = s;
    __syncthreads();
    for (int off = 128; off >= 1; off >>= 1) {
        if ((int)threadIdx.x < off) sh[threadIdx.x] += sh[threadIdx.x + off];
        __syncthreads();
    }
    if (threadIdx.x == 0) {
        const float scale = 1.0f / ((float)BATCH * (float)NPTS * (float)NPTS);
        out[0] = sh[0] * scale;
    }
}

// ---------------------------------------------------------------------------
extern "C" void kernel_launch(void* const* d_in, const int* in_sizes, int n_in,
                              void* d_out, int out_size, void* d_ws, size_t ws_size,
                              hipStream_t stream) {
    (void)in_sizes; (void)n_in; (void)out_size; (void)ws_size;

    const float* xo = (const float*)d_in[0];   // output  [8,2048,64] f32
    const float* xt = (const float*)d_in[1];   // target  [8,2048,64] f32

    float* sqo      = (float*)d_ws;            //  65536 B
    float* sqt      = sqo + BATCH * NPTS;      //  65536 B
    float* partials = sqt + BATCH * NPTS;      //   4096 B   (total ~136 KB)

    norms_kernel<<<(BATCH * NPTS + 255) / 256, 256, 0, stream>>>(xo, xt, sqo, sqt);
    dist_tile_kernel<<<BATCH * NT, 256, 0, stream>>>(xo, xt, sqo, sqt, partials);
    reduce_kernel<<<1, 256, 0, stream>>>(partials, (float*)d_out);
}